// NGCF_28965259444540
// MI455X (gfx1250) — compile-verified
//
#include <hip/hip_runtime.h>
#include <hip/hip_bf16.h>
#include <math.h>

typedef float v2f __attribute__((ext_vector_type(2)));
typedef float v8f __attribute__((ext_vector_type(8)));

#define NNODES    150000
#define EMBD      64
#define NLAYERS   3
#define NEG_SLOPE 0.01f

// ---------------------------------------------------------------------------
// init: ego = emb ; out[:, 0:64] = emb   (row stride of out is 256)
// ---------------------------------------------------------------------------
__global__ __launch_bounds__(256) void ngcf_init(const float* __restrict__ emb,
                                                 float* __restrict__ ego,
                                                 float* __restrict__ out) {
  int idx = blockIdx.x * 256 + threadIdx.x;           // 0 .. N*16-1
  if (idx >= NNODES * (EMBD / 4)) return;
  int row = idx >> 4;
  int c4  = (idx & 15) * 4;
  float4 v = *(const float4*)(emb + (size_t)row * EMBD + c4);
  *(float4*)(ego + (size_t)row * EMBD + c4) = v;
  *(float4*)(out + (size_t)row * (4 * EMBD) + c4) = v;
}

// ---------------------------------------------------------------------------
// zero the SpMM accumulator
// ---------------------------------------------------------------------------
__global__ __launch_bounds__(256) void ngcf_zero(float* __restrict__ side) {
  int idx = blockIdx.x * 256 + threadIdx.x;
  if (idx >= NNODES * (EMBD / 4)) return;
  *(float4*)(side + (size_t)idx * 4) = make_float4(0.f, 0.f, 0.f, 0.f);
}

// ---------------------------------------------------------------------------
// SpMM: side[rows[e]] += vals[e] * ego[cols[e]]
// 16 lanes per edge, float4 per lane, hardware FP32 atomics.
// ---------------------------------------------------------------------------
__global__ __launch_bounds__(256) void ngcf_spmm(const int* __restrict__ rows,
                                                 const int* __restrict__ cols,
                                                 const float* __restrict__ vals,
                                                 const float* __restrict__ ego,
                                                 float* __restrict__ side,
                                                 int nEdges) {
  int tid = blockIdx.x * 256 + threadIdx.x;
  int e = tid >> 4;
  if (e >= nEdges) return;
  int j = (tid & 15) * 4;
  int r = rows[e];
  int c = cols[e];
  float v = vals[e];
  float4 x = *(const float4*)(ego + (size_t)c * EMBD + j);
  float* dst = side + (size_t)r * EMBD + j;
#if defined(__HIP_DEVICE_COMPILE__)
  unsafeAtomicAdd(dst + 0, v * x.x);
  unsafeAtomicAdd(dst + 1, v * x.y);
  unsafeAtomicAdd(dst + 2, v * x.z);
  unsafeAtomicAdd(dst + 3, v * x.w);
#endif
}

// ---------------------------------------------------------------------------
// Fused dense layer with fragment-swizzled weights in LDS:
//   sum = leaky(side@W1 + b1); bi = leaky((ego*side)@W2 + b2)
//   ego_new = sum + bi (in-place); out[:, 64*(l+1)] = ego_new / ||row||
// 8 waves/block, one 16-row tile per wave, V_WMMA_F32_16X16X4_F32.
// LDS layout: lW[(kb*4+nt)*64 + lane*2 + j] = W[kb*4 + 2*(lane>=16) + j][nt*16 + lane&15]
// so each B fragment is ONE conflict-free ds_load_b64 into an aligned pair.
// ---------------------------------------------------------------------------
__global__ __launch_bounds__(256) void ngcf_dense(const float* __restrict__ side,
                                                  float* __restrict__ ego,
                                                  const float* __restrict__ W1,
                                                  const float* __restrict__ b1,
                                                  const float* __restrict__ W2,
                                                  const float* __restrict__ b2,
                                                  float* __restrict__ out,
                                                  int layer) {
  __shared__ float lW1[EMBD * EMBD];   // fragment-major swizzle
  __shared__ float lW2[EMBD * EMBD];
  __shared__ float lb1[EMBD];
  __shared__ float lb2[EMBD];

  const float* W1l = W1 + (size_t)layer * EMBD * EMBD;
  const float* W2l = W2 + (size_t)layer * EMBD * EMBD;
  for (int idx = threadIdx.x; idx < EMBD * EMBD; idx += 256) {
    const int kbnt = idx >> 6;          // fragment id: kb*4 + nt
    const int l    = (idx >> 1) & 31;   // destination lane
    const int j    = idx & 1;
    const int kb   = kbnt >> 2;
    const int nt   = kbnt & 3;
    const int k    = kb * 4 + ((l >> 4) << 1) + j;
    const int n    = nt * 16 + (l & 15);
    lW1[idx] = W1l[k * EMBD + n];
    lW2[idx] = W2l[k * EMBD + n];
  }
  if (threadIdx.x < EMBD) {
    lb1[threadIdx.x] = b1[(size_t)layer * EMBD + threadIdx.x];
    lb2[threadIdx.x] = b2[(size_t)layer * EMBD + threadIdx.x];
  }
  __syncthreads();

  const int wave = threadIdx.x >> 5;
  const int lane = threadIdx.x & 31;
  const int tileBase = blockIdx.x * 128 + wave * 16;
  if (tileBase >= NNODES) return;            // wave-uniform: EXEC stays all-ones

  const int lh  = lane >> 4;                 // lane half (0 / 1)
  const int ln  = lane & 15;
  const int klo = lh * 2;                    // K sub-offset per fragment layout

  const float* srow = side + (size_t)(tileBase + ln) * EMBD;
  const float* erow = ego  + (size_t)(tileBase + ln) * EMBD;
  const float* fw1  = lW1 + lane * 2;        // per-lane fragment base
  const float* fw2  = lW2 + lane * 2;

  v8f acc1[4];
  v8f acc2[4];
#pragma unroll
  for (int nt = 0; nt < 4; ++nt) { acc1[nt] = (v8f){}; acc2[nt] = (v8f){}; }

#pragma unroll
  for (int kb = 0; kb < 16; ++kb) {
    const int k = kb * 4 + klo;
    v2f aS = *(const v2f*)(srow + k);        // global_load_b64
    v2f aE = *(const v2f*)(erow + k);
    v2f aB = {aE.x * aS.x, aE.y * aS.y};     // v_pk_mul_f32
#pragma unroll
    for (int nt = 0; nt < 4; ++nt) {
      const int fo = (kb * 4 + nt) * EMBD;   // fragment offset in LDS
      v2f bA = *(const v2f*)(fw1 + fo);      // one ds_load_b64, conflict-free
      v2f bB = *(const v2f*)(fw2 + fo);
      acc1[nt] = __builtin_amdgcn_wmma_f32_16x16x4_f32(
          false, aS, false, bA, (short)0, acc1[nt], false, false);
      acc2[nt] = __builtin_amdgcn_wmma_f32_16x16x4_f32(
          false, aB, false, bB, (short)0, acc2[nt], false, false);
    }
  }

  // bias + leaky relu + add
  float ev[4][8];
#pragma unroll
  for (int nt = 0; nt < 4; ++nt) {
    const int n = nt * 16 + ln;
    const float bb1 = lb1[n];
    const float bb2 = lb2[n];
#pragma unroll
    for (int r = 0; r < 8; ++r) {
      float s = acc1[nt][r] + bb1;
      s = (s >= 0.f) ? s : NEG_SLOPE * s;
      float t = acc2[nt][r] + bb2;
      t = (t >= 0.f) ? t : NEG_SLOPE * t;
      ev[nt][r] = s + t;
    }
  }

  // per-row L2 norm: in-lane sum over 4 n-tiles, then 16-lane butterfly.
  const int mbase = tileBase + lh * 8;
  float* outCol = out + (size_t)(layer + 1) * EMBD;
#pragma unroll
  for (int r = 0; r < 8; ++r) {
    float ss = ev[0][r] * ev[0][r] + ev[1][r] * ev[1][r] +
               ev[2][r] * ev[2][r] + ev[3][r] * ev[3][r];
    ss += __shfl_xor(ss, 1, 16);
    ss += __shfl_xor(ss, 2, 16);
    ss += __shfl_xor(ss, 4, 16);
    ss += __shfl_xor(ss, 8, 16);
    const float inv = 1.0f / fmaxf(sqrtf(ss), 1e-12f);
    const int m = mbase + r;
#pragma unroll
    for (int nt = 0; nt < 4; ++nt) {
      const int n = nt * 16 + ln;
      ego[(size_t)m * EMBD + n] = ev[nt][r];
      outCol[(size_t)m * (4 * EMBD) + n] = ev[nt][r] * inv;
    }
  }
}

// ---------------------------------------------------------------------------
extern "C" void kernel_launch(void* const* d_in, const int* in_sizes, int n_in,
                              void* d_out, int out_size, void* d_ws, size_t ws_size,
                              hipStream_t stream) {
  const float* emb  = (const float*)d_in[0];
  const int*   rows = (const int*)d_in[1];
  const int*   cols = (const int*)d_in[2];
  const float* vals = (const float*)d_in[3];
  const float* W1   = (const float*)d_in[4];
  const float* b1   = (const float*)d_in[5];
  const float* W2   = (const float*)d_in[6];
  const float* b2   = (const float*)d_in[7];
  float* out = (float*)d_out;
  const int nEdges = in_sizes[1];

  float* ego  = (float*)d_ws;                               // N*64 floats
  float* side = (float*)d_ws + (size_t)NNODES * EMBD;       // N*64 floats

  const int vecBlocks   = (NNODES * (EMBD / 4) + 255) / 256;        // 9375
  const int spmmBlocks  = ((nEdges * 16) + 255) / 256;
  const int denseBlocks = (NNODES + 127) / 128;                     // 1172

  ngcf_init<<<vecBlocks, 256, 0, stream>>>(emb, ego, out);

  for (int l = 0; l < NLAYERS; ++l) {
    ngcf_zero<<<vecBlocks, 256, 0, stream>>>(side);
    ngcf_spmm<<<spmmBlocks, 256, 0, stream>>>(rows, cols, vals, ego, side, nEdges);
    ngcf_dense<<<denseBlocks, 256, 0, stream>>>(side, ego, W1, b1, W2, b2, out, l);
  }
}